// GraphDualModel_54193897341273
// MI455X (gfx1250) — compile-verified
//
#include <hip/hip_runtime.h>
#include <hip/hip_bf16.h>
#include <math.h>

#define N_NODES  1024
#define N_EDGES  32768
#define FLAT_DIM 4096   // 4*N

typedef __attribute__((ext_vector_type(2))) float v2f;
typedef __attribute__((ext_vector_type(8))) float v8f;

// ---------------------------------------------------------------------------
// ws layout (floats): [0, 4096) agg/flat ; [4096, 4096+32768) policy logits
// ---------------------------------------------------------------------------
__global__ void k_init(const float* __restrict__ bp, float* __restrict__ ws) {
  int i = blockIdx.x * blockDim.x + threadIdx.x;
  if (i < FLAT_DIM) ws[i] = 0.0f;
  if (i < N_EDGES)  ws[FLAT_DIM + i] = bp[i];
}

// ---------------------------------------------------------------------------
// EdgeConv: one thread per edge. One-hot algebra -> 3 gathered rows of w1,
// 10->4 MLP, 4-way softmax, atomic scatter-add into agg[dst][0..3].
// ---------------------------------------------------------------------------
__global__ void k_edge_mlp(const int* __restrict__ tgt,
                           const int* __restrict__ eidx,
                           const float* __restrict__ w1, const float* __restrict__ b1,
                           const float* __restrict__ w2, const float* __restrict__ b2,
                           float* __restrict__ agg) {
  int e = blockIdx.x * blockDim.x + threadIdx.x;
  if (e >= N_EDGES) return;
  int src = eidx[e];
  int dst = eidx[N_EDGES + e];
  int td = tgt[dst];
  int ts = tgt[src];

  float h[10];
#pragma unroll
  for (int j = 0; j < 10; ++j) h[j] = b1[j];
  if (td >= 0) {
    const float* r0 = w1 + (long)td * 10;               // xi @ w1[:N]
    const float* r1 = w1 + (long)(N_NODES + td) * 10;   // -xi @ w1[N:]
#pragma unroll
    for (int j = 0; j < 10; ++j) h[j] += r0[j] - r1[j];
  }
  if (ts >= 0) {
    const float* r2 = w1 + (long)(N_NODES + ts) * 10;   // xj @ w1[N:]
#pragma unroll
    for (int j = 0; j < 10; ++j) h[j] += r2[j];
  }

  float z[4];
#pragma unroll
  for (int c = 0; c < 4; ++c) z[c] = b2[c];
#pragma unroll
  for (int j = 0; j < 10; ++j) {
    float hj = fmaxf(h[j], 0.0f);                       // ReLU
#pragma unroll
    for (int c = 0; c < 4; ++c) z[c] += hj * w2[j * 4 + c];
  }
  float m = fmaxf(fmaxf(z[0], z[1]), fmaxf(z[2], z[3]));
  float ez0 = __expf(z[0] - m), ez1 = __expf(z[1] - m);
  float ez2 = __expf(z[2] - m), ez3 = __expf(z[3] - m);
  float inv = 1.0f / (ez0 + ez1 + ez2 + ez3);
  float* row = agg + (long)dst * 4;
  atomicAdd(row + 0, ez0 * inv);
  atomicAdd(row + 1, ez1 * inv);
  atomicAdd(row + 2, ez2 * inv);
  atomicAdd(row + 3, ez3 * inv);
}

// ---------------------------------------------------------------------------
// Policy GEMV via V_WMMA_F32_16X16X4_F32.
//   D[m,n] += sum_k A[m,k]*B[k,n]; A rows all = flat[kb..kb+3] (broadcast),
//   B[k,n] = wp[(kb+k)*E + colbase + n]. All D rows equal the 16 logits.
// Layouts (ISA 7.12.2): A: lanes0-15 M, VGPR0/1 = K0/K1; lanes16-31 = K2/K3.
//   B mirrors with lane = N. D VGPR0, lanes0-15 = row M=0.
// Wave w: tile = w>>2 (16 cols), split-K quadrant = w&3 (1024 K each).
// 8192 waves, fully coalesced 64B wp row segments, 512MB streamed once.
// ---------------------------------------------------------------------------
__global__ void k_policy_gemv(const float* __restrict__ wp,
                              const float* __restrict__ flat,
                              float* __restrict__ logits) {
  const int wave = (blockIdx.x * blockDim.x + threadIdx.x) >> 5;
  const int lane = threadIdx.x & 31;
  const int tile = wave >> 2;           // 0..2047
  const int ks   = wave & 3;            // split-K quadrant
  const int colbase = tile * 16;
  const int half = lane >> 4;           // selects K pair {0,1} or {2,3}
  const int n    = lane & 15;
  const float* colp = wp + colbase + n;
  const int k0 = ks * 1024;

  v8f acc = {};
#pragma unroll 4
  for (int kb = k0; kb < k0 + 1024; kb += 4) {
    int ka = kb + half * 2;
    v2f a, b;
    a.x = flat[ka];
    a.y = flat[ka + 1];
    b.x = colp[(long)ka * N_EDGES];
    b.y = colp[(long)(ka + 1) * N_EDGES];
    acc = __builtin_amdgcn_wmma_f32_16x16x4_f32(
        /*neg_a=*/false, a, /*neg_b=*/false, b,
        /*c_mod=*/(short)0, acc, /*reuse_a=*/false, /*reuse_b=*/false);
  }
  if (lane < 16) atomicAdd(&logits[colbase + n], acc[0]);   // D row M=0
}

// ---------------------------------------------------------------------------
// Softmax over 32768 logits: one 1024-thread block, wave32 shuffle reductions.
// ---------------------------------------------------------------------------
__global__ void k_softmax(const float* __restrict__ logits, float* __restrict__ out) {
  __shared__ float red[32];
  __shared__ float bval;
  const int tid = threadIdx.x, wid = tid >> 5, lane = tid & 31;

  float m = -3.402823466e38f;
  for (int i = tid; i < N_EDGES; i += 1024) m = fmaxf(m, logits[i]);
#pragma unroll
  for (int off = 16; off; off >>= 1) m = fmaxf(m, __shfl_xor(m, off, 32));
  if (lane == 0) red[wid] = m;
  __syncthreads();
  if (wid == 0) {
    float v = red[lane];
#pragma unroll
    for (int off = 16; off; off >>= 1) v = fmaxf(v, __shfl_xor(v, off, 32));
    if (lane == 0) bval = v;
  }
  __syncthreads();
  const float M = bval;

  float s = 0.0f;
  for (int i = tid; i < N_EDGES; i += 1024) s += __expf(logits[i] - M);
#pragma unroll
  for (int off = 16; off; off >>= 1) s += __shfl_xor(s, off, 32);
  __syncthreads();
  if (lane == 0) red[wid] = s;
  __syncthreads();
  if (wid == 0) {
    float v = red[lane];
#pragma unroll
    for (int off = 16; off; off >>= 1) v += __shfl_xor(v, off, 32);
    if (lane == 0) bval = v;
  }
  __syncthreads();
  const float inv = 1.0f / bval;
  for (int i = tid; i < N_EDGES; i += 1024)
    out[i] = __expf(logits[i] - M) * inv;
}

// ---------------------------------------------------------------------------
// Value head: 4096->64->32->16->1, single 64-thread block (1 MB of wv1).
// ---------------------------------------------------------------------------
__global__ void k_value(const float* __restrict__ flat,
                        const float* __restrict__ wv1, const float* __restrict__ bv1,
                        const float* __restrict__ wv2, const float* __restrict__ bv2,
                        const float* __restrict__ wv3, const float* __restrict__ bv3,
                        const float* __restrict__ wv4, const float* __restrict__ bv4,
                        float* __restrict__ out) {
  __shared__ float v1[64];
  __shared__ float v2s[32];
  __shared__ float v3s[16];
  const int t = threadIdx.x;   // 64 threads

  float s = bv1[t];
  for (int k = 0; k < FLAT_DIM; ++k) s += flat[k] * wv1[(long)k * 64 + t];
  v1[t] = fmaxf(s, 0.0f);
  __syncthreads();

  if (t < 32) {
    s = bv2[t];
    for (int k = 0; k < 64; ++k) s += v1[k] * wv2[k * 32 + t];
    v2s[t] = fmaxf(s, 0.0f);
  }
  __syncthreads();

  if (t < 16) {
    s = bv3[t];
    for (int k = 0; k < 32; ++k) s += v2s[k] * wv3[k * 16 + t];
    v3s[t] = fmaxf(s, 0.0f);
  }
  __syncthreads();

  if (t == 0) {
    s = bv4[0];
    for (int k = 0; k < 16; ++k) s += v3s[k] * wv4[k];
    out[N_EDGES] = s;
  }
}

// ---------------------------------------------------------------------------
extern "C" void kernel_launch(void* const* d_in, const int* in_sizes, int n_in,
                              void* d_out, int out_size, void* d_ws, size_t ws_size,
                              hipStream_t stream) {
  const int*   tgt  = (const int*)d_in[0];
  const int*   eidx = (const int*)d_in[1];
  const float* w1   = (const float*)d_in[2];
  const float* b1   = (const float*)d_in[3];
  const float* w2   = (const float*)d_in[4];
  const float* b2   = (const float*)d_in[5];
  const float* wv1  = (const float*)d_in[6];
  const float* bv1  = (const float*)d_in[7];
  const float* wv2  = (const float*)d_in[8];
  const float* bv2  = (const float*)d_in[9];
  const float* wv3  = (const float*)d_in[10];
  const float* bv3  = (const float*)d_in[11];
  const float* wv4  = (const float*)d_in[12];
  const float* bv4  = (const float*)d_in[13];
  const float* wp   = (const float*)d_in[14];
  const float* bp   = (const float*)d_in[15];

  float* out    = (float*)d_out;
  float* ws     = (float*)d_ws;
  float* agg    = ws;              // 4096 floats == flat
  float* logits = ws + FLAT_DIM;   // 32768 floats

  k_init<<<N_EDGES / 256, 256, 0, stream>>>(bp, ws);
  k_edge_mlp<<<N_EDGES / 256, 256, 0, stream>>>(tgt, eidx, w1, b1, w2, b2, agg);
  // 2048 column tiles * 4 split-K waves = 8192 waves = 1024 blocks of 256
  k_policy_gemv<<<1024, 256, 0, stream>>>(wp, agg, logits);
  k_softmax<<<1, 1024, 0, stream>>>(logits, out);
  k_value<<<1, 64, 0, stream>>>(agg, wv1, bv1, wv2, bv2, wv3, bv3, wv4, bv4, out);
}